// HHEdgeCons_16922171147095
// MI455X (gfx1250) — compile-verified
//
#include <hip/hip_runtime.h>
#include <math.h>

typedef float v2f __attribute__((ext_vector_type(2)));
typedef float v8f __attribute__((ext_vector_type(8)));

#define NN 4096
#define TT 4
#define FF 64
#define M_TOTAL (NN * TT)   // 16384 rows of lin_sq
#define K_TOTAL NN          // 4096
#define BK 128              // K-chunk per LDS stage
#define BKP 132             // padded row stride (bank-conflict-free, 16B aligned)

// ---------------------------------------------------------------------------
// ws[0] = sum |lin_sq|   (L1)
// ws[1] = sum_r ||lin_sq_r||2
// ws[2] = sum_r ||diff_r||2
// ---------------------------------------------------------------------------
__global__ void init_ws_kernel(float* ws) {
    if (threadIdx.x < 3) ws[threadIdx.x] = 0.0f;
}

__global__ __launch_bounds__(128)
void fused_relu_gemm_kernel(const float* __restrict__ feature,   // [4096,64]
                            const float* __restrict__ mask,      // [16384,4096] flat
                            const float* __restrict__ linear,    // [16384,4096] flat
                            const float* __restrict__ proj,      // [16,64,64]
                            const int*   __restrict__ nmm,       // [4096]
                            float*       __restrict__ out_lin,   // [16384,4096] flat
                            float*       __restrict__ ws)
{
    __shared__ __align__(16) float At[16][BKP];   // lin_sq tile (16 x BK, padded)
    __shared__ __align__(16) float Ft[BK][FF];    // feature chunk (BK x 64)
    __shared__ float rowSumSq[16];
    __shared__ float rowDiffSq[16];
    __shared__ float l1s;

    const int tid  = threadIdx.x;
    const int r0   = blockIdx.x * 16;      // first global row of this tile
    const int wave = tid >> 5;             // 0..3 -> column tile
    const int lane = tid & 31;
    const int half = lane >> 4;            // 0 or 1
    const int l15  = lane & 15;
    const int c0   = wave * 16;            // first column of this wave's tile

    // phase-A assignment: thread covers row (tid>>3), 16 contiguous cols
    const int arow = tid >> 3;             // 0..15
    const int aseg = (tid & 7) * 16;       // 0..112

    if (tid < 16) { rowSumSq[tid] = 0.0f; rowDiffSq[tid] = 0.0f; }
    if (tid == 0) l1s = 0.0f;

    float l1_local = 0.0f;
    float sq_local = 0.0f;
    v8f   c = {};                          // f32 WMMA accumulator (16x16 tile)

    for (int k0 = 0; k0 < K_TOTAL; k0 += BK) {
        __syncthreads();   // LDS reuse guard (also covers initial stores)

        // ---- stage feature chunk [k0 .. k0+BK) x 64 into LDS (coalesced) ----
        {
            const float4* fsrc = (const float4*)feature + (size_t)k0 * (FF / 4);
            float4* fdst = (float4*)&Ft[0][0];
            #pragma unroll
            for (int j = 0; j < 16; ++j)
                fdst[tid + j * 128] = fsrc[tid + j * 128];
        }

        // ---- elementwise relu(linear*mask): write output + stage A tile ----
        {
            const size_t base = (size_t)(r0 + arow) * K_TOTAL + k0 + aseg;
            #pragma unroll
            for (int i = 0; i < 4; ++i) {
                float4 lv = *(const float4*)(linear + base + 4 * i);
                float4 mv = *(const float4*)(mask   + base + 4 * i);
                float4 s;
                s.x = lv.x * mv.x; s.x = (s.x > 0.0f) ? s.x : 0.0f;
                s.y = lv.y * mv.y; s.y = (s.y > 0.0f) ? s.y : 0.0f;
                s.z = lv.z * mv.z; s.z = (s.z > 0.0f) ? s.z : 0.0f;
                s.w = lv.w * mv.w; s.w = (s.w > 0.0f) ? s.w : 0.0f;
                *(float4*)(out_lin + base + 4 * i) = s;
                *(float4*)&At[arow][aseg + 4 * i] = s;
                l1_local += s.x + s.y + s.z + s.w;
                sq_local += s.x * s.x + s.y * s.y + s.z * s.z + s.w * s.w;
            }
        }
        __syncthreads();

        // ---- FP32 WMMA sweep over the chunk: C += A(16x4) * B(4x16) ----
        // A frag: lanes 0-15 -> M=lane, K=kk/kk+1 ; lanes 16-31 -> M=lane-16, K=kk+2/kk+3
        // B frag: lanes 0-15 -> N=lane, K=kk/kk+1 ; lanes 16-31 -> N=lane-16, K=kk+2/kk+3
        #pragma unroll 8
        for (int kk = 0; kk < BK; kk += 4) {
            v2f a = *(const v2f*)&At[l15][kk + 2 * half];
            v2f b;
            b.x = Ft[kk + 2 * half + 0][c0 + l15];
            b.y = Ft[kk + 2 * half + 1][c0 + l15];
            c = __builtin_amdgcn_wmma_f32_16x16x4_f32(
                    /*neg_a=*/false, a, /*neg_b=*/false, b,
                    /*c_mod=*/(short)0, c, /*reuse_a=*/false, /*reuse_b=*/false);
        }
    }

    // ---- per-row sum-of-squares + L1 reduction ----
    atomicAdd(&rowSumSq[arow], sq_local);
    atomicAdd(&l1s, l1_local);
    __syncthreads();

    // ---- inp_projected for this wave's 16x16 C-tile, diff norms ----
    // D layout: vgpr j, lane -> element (row = j + 8*half, col = l15)
    #pragma unroll
    for (int j = 0; j < 8; ++j) {
        const int tr  = j + 8 * half;        // row within tile (0..15)
        const int gr  = r0 + tr;             // global row = n*T + t
        const int n   = gr >> 2;
        const int t   = gr & 3;
        const int col = c0 + l15;
        const int pidx = nmm[n] * TT + t;
        const float4* prow = (const float4*)(proj + (size_t)pidx * FF * FF + (size_t)col * FF);
        const float4* frow = (const float4*)(feature + (size_t)n * FF);
        float acc = 0.0f;
        #pragma unroll
        for (int g = 0; g < 16; ++g) {
            float4 p = prow[g];
            float4 f = frow[g];
            acc += p.x * f.x + p.y * f.y + p.z * f.z + p.w * f.w;
        }
        const float d = acc - c[j];
        atomicAdd(&rowDiffSq[tr], d * d);
    }
    __syncthreads();

    // ---- workgroup partials -> global accumulators ----
    if (tid == 0) {
        float l2p = 0.0f, rcp = 0.0f;
        #pragma unroll
        for (int i = 0; i < 16; ++i) {
            l2p += sqrtf(rowSumSq[i]);
            rcp += sqrtf(rowDiffSq[i]);
        }
        atomicAdd(&ws[0], l1s);
        atomicAdd(&ws[1], l2p);
        atomicAdd(&ws[2], rcp);
    }
}

__global__ void finalize_kernel(const float* __restrict__ ws,
                                float* __restrict__ out_scalar)
{
    if (threadIdx.x == 0 && blockIdx.x == 0) {
        const float recon_loss = 0.01f * ws[2] + 0.2f * ws[1] + ws[0];
        *out_scalar = 0.1f * recon_loss;   // RECONS_ERROR_LAMBDA * recon_loss
    }
}

extern "C" void kernel_launch(void* const* d_in, const int* in_sizes, int n_in,
                              void* d_out, int out_size, void* d_ws, size_t ws_size,
                              hipStream_t stream)
{
    (void)in_sizes; (void)n_in; (void)out_size; (void)ws_size;
    const float* feature = (const float*)d_in[0];
    const float* mask    = (const float*)d_in[1];
    const float* linear  = (const float*)d_in[2];
    const float* proj    = (const float*)d_in[3];
    const int*   nmm     = (const int*)d_in[4];
    float* out = (float*)d_out;
    float* ws  = (float*)d_ws;

    hipLaunchKernelGGL(init_ws_kernel, dim3(1), dim3(32), 0, stream, ws);
    hipLaunchKernelGGL(fused_relu_gemm_kernel, dim3(M_TOTAL / 16), dim3(128), 0, stream,
                       feature, mask, linear, proj, nmm, out, ws);
    hipLaunchKernelGGL(finalize_kernel, dim3(1), dim3(1), 0, stream,
                       ws, out + (size_t)M_TOTAL * K_TOTAL);
}